// Actor_GRU_56727928046063
// MI455X (gfx1250) — compile-verified
//
#include <hip/hip_runtime.h>
#include <hip/hip_bf16.h>
#include <math.h>

// Problem constants (from reference)
#define TT   32768        // timesteps
#define DD   256          // input feature dim (states has D+3 = 259 cols)
#define SDD  259          // states row stride
#define GG   192          // 3*H gate dim
#define HH   64           // hidden dim
#define NOUT 1024         // J*J output dim

typedef __attribute__((ext_vector_type(2))) float v2f;
typedef __attribute__((ext_vector_type(8))) float v8f;

// ---------------------------------------------------------------------------
// Kernel 1: gi = X @ W_ih^T + b_ih   (T x 192), fp32 WMMA 16x16x4
// One wave per 16x16 output tile; K loop over 256 in steps of 4.
// ---------------------------------------------------------------------------
__global__ __launch_bounds__(256) void gemm_gi_kernel(
    const float* __restrict__ states,   // (T, 259); X = cols [0,256)
    const float* __restrict__ w_ih,     // (192, 256) row-major
    const float* __restrict__ b_ih,     // (192,)
    float* __restrict__ gi)             // (T, 192) row-major
{
    const int NT = GG / 16;                               // 12 tiles along N
    int wave = (blockIdx.x << 3) + (threadIdx.x >> 5);    // 8 waves / block
    int mt = wave / NT;
    int nt = wave % NT;
    int lane = threadIdx.x & 31;
    int half = lane >> 4;
    int idx  = lane & 15;

    int m = mt * 16 + idx;            // A row for this lane
    int n = nt * 16 + idx;            // B/C column for this lane

    const float* arow = states + (size_t)m * SDD;     // stride 259 (4B aligned only)
    const float* brow = w_ih   + (size_t)n * DD;      // stride 256 (8B aligned)

    v8f acc = {};
#pragma unroll 8
    for (int k = 0; k < DD; k += 4) {
        v2f a, b;
        int ka = k + 2 * half;
        a.x = arow[ka];
        a.y = arow[ka + 1];
        b = *(const v2f*)(brow + ka);                 // contiguous, 8B aligned
        acc = __builtin_amdgcn_wmma_f32_16x16x4_f32(
            false, a, false, b, (short)0, acc, false, false);
    }

    float bias = b_ih[n];
#pragma unroll
    for (int r = 0; r < 8; ++r) {
        int row = mt * 16 + r + 8 * half;
        gi[(size_t)row * GG + n] = acc[r] + bias;
    }
}

// ---------------------------------------------------------------------------
// Kernel 2: sequential GRU scan. 1 block, 64 lanes (2 waves).
// Lane d owns h[d] and W_hh rows {d, 64+d, 128+d} in registers (192 VGPRs).
// h broadcast via LDS double buffer -> exactly one barrier per step.
// ---------------------------------------------------------------------------
__device__ __forceinline__ float fast_sigmoid(float x) {
    return 1.0f / (1.0f + __expf(-x));
}
__device__ __forceinline__ float fast_tanh(float x) {
    // 1 - 2/(e^{2x}+1): saturates correctly to +/-1 without inf/inf
    float e = __expf(2.0f * x);
    return 1.0f - 2.0f / (e + 1.0f);
}

__global__ __launch_bounds__(64, 1) void gru_scan_kernel(
    const float* __restrict__ gi,      // (T, 192)
    const float* __restrict__ w_hh,    // (192, 64) row-major
    const float* __restrict__ b_hh,    // (192,)
    const float* __restrict__ h0,      // (64,)
    float* __restrict__ hs,            // (T, 64)
    float* __restrict__ h_final)       // (64,)
{
    const int d = threadIdx.x;         // 0..63
    __shared__ float hbuf[2][HH];

    // Persistent per-lane weights (registers; fully unrolled access below)
    float wr[HH], wz[HH], wn[HH];
#pragma unroll
    for (int k = 0; k < HH; ++k) {
        wr[k] = w_hh[(size_t)d * HH + k];
        wz[k] = w_hh[(size_t)(HH + d) * HH + k];
        wn[k] = w_hh[(size_t)(2 * HH + d) * HH + k];
    }
    const float br = b_hh[d];
    const float bz = b_hh[HH + d];
    const float bn = b_hh[2 * HH + d];

    float h = h0[d];
    hbuf[0][d] = h;
    __syncthreads();

    int cur = 0;
    const float* gp = gi;
    float* hp = hs;
    for (int t = 0; t < TT; ++t) {
        // gi for this step (no dependence on h; latency hidden under FMAs)
        float ir = gp[d];
        float iz = gp[HH + d];
        float inn = gp[2 * HH + d];
        gp += GG;

        float ar = br, az = bz, an = bn;
#pragma unroll
        for (int k = 0; k < HH; ++k) {
            float hv = hbuf[cur][k];   // LDS broadcast read
            ar = fmaf(wr[k], hv, ar);
            az = fmaf(wz[k], hv, az);
            an = fmaf(wn[k], hv, an);
        }

        float r = fast_sigmoid(ir + ar);
        float z = fast_sigmoid(iz + az);
        float nn = fast_tanh(inn + r * an);
        h = (1.0f - z) * nn + z * h;

        hp[d] = h;                     // fire-and-forget global store
        hp += HH;
        hbuf[cur ^ 1][d] = h;          // write other buffer (no race w/ readers)
        __syncthreads();               // one barrier per step
        cur ^= 1;
    }
    h_final[d] = h;
}

// ---------------------------------------------------------------------------
// Kernel 3: logits = HS @ W_out^T + b_out, masked.  fp32 WMMA, K=64.
// Mask depends only on the output column == per-lane N index.
// ---------------------------------------------------------------------------
__global__ __launch_bounds__(256) void gemm_out_kernel(
    const float* __restrict__ hs,      // (T, 64)
    const float* __restrict__ w_out,   // (1024, 64) row-major
    const float* __restrict__ b_out,   // (1024,)
    const int*   __restrict__ mask,    // (32,32) -> flat 1024
    float* __restrict__ logits)        // (T, 1024)
{
    const int NT = NOUT / 16;                             // 64 tiles along N
    int wave = (blockIdx.x << 3) + (threadIdx.x >> 5);
    int mt = wave / NT;
    int nt = wave % NT;
    int lane = threadIdx.x & 31;
    int half = lane >> 4;
    int idx  = lane & 15;

    int m = mt * 16 + idx;
    int n = nt * 16 + idx;

    const float* arow = hs    + (size_t)m * HH;           // stride 64, 8B aligned
    const float* brow = w_out + (size_t)n * HH;           // stride 64, 8B aligned

    v8f acc = {};
#pragma unroll
    for (int k = 0; k < HH; k += 4) {
        int ka = k + 2 * half;
        v2f a = *(const v2f*)(arow + ka);
        v2f b = *(const v2f*)(brow + ka);
        acc = __builtin_amdgcn_wmma_f32_16x16x4_f32(
            false, a, false, b, (short)0, acc, false, false);
    }

    float bias = b_out[n];
    bool masked = (mask[n] == 0);
#pragma unroll
    for (int r = 0; r < 8; ++r) {
        int row = mt * 16 + r + 8 * half;
        float v = acc[r] + bias;
        logits[(size_t)row * NOUT + n] = masked ? -1000000000.0f : v;
    }
}

// ---------------------------------------------------------------------------
// Launch wiring.
// Inputs (setup_inputs order): states, h, w_ih, w_hh, b_ih, b_hh, w_out,
//                              b_out, mask
// Output: logits (T*1024 f32) || h_final (64 f32)
// ---------------------------------------------------------------------------
extern "C" void kernel_launch(void* const* d_in, const int* in_sizes, int n_in,
                              void* d_out, int out_size, void* d_ws, size_t ws_size,
                              hipStream_t stream) {
    const float* states = (const float*)d_in[0];
    const float* h0     = (const float*)d_in[1];
    const float* w_ih   = (const float*)d_in[2];
    const float* w_hh   = (const float*)d_in[3];
    const float* b_ih   = (const float*)d_in[4];
    const float* b_hh   = (const float*)d_in[5];
    const float* w_out  = (const float*)d_in[6];
    const float* b_out  = (const float*)d_in[7];
    const int*   mask   = (const int*)d_in[8];

    float* logits  = (float*)d_out;                       // T*1024
    float* h_final = (float*)d_out + (size_t)TT * NOUT;   // 64

    // Workspace: gi (T*192 f32) then hs (T*64 f32) = ~33.6 MB
    float* gi = (float*)d_ws;
    float* hs = gi + (size_t)TT * GG;

    // GEMM1: (T/16)*(192/16) = 2048*12 = 24576 tiles; 8 waves per block
    {
        int tiles = (TT / 16) * (GG / 16);
        dim3 grid(tiles / 8), block(256);
        gemm_gi_kernel<<<grid, block, 0, stream>>>(states, w_ih, b_ih, gi);
    }
    // Sequential scan: single workgroup of 64 lanes
    {
        gru_scan_kernel<<<dim3(1), dim3(64), 0, stream>>>(gi, w_hh, b_hh, h0,
                                                          hs, h_final);
    }
    // GEMM3: 2048*64 = 131072 tiles; 8 waves per block
    {
        int tiles = (TT / 16) * (NOUT / 16);
        dim3 grid(tiles / 8), block(256);
        gemm_out_kernel<<<grid, block, 0, stream>>>(hs, w_out, b_out, mask,
                                                    logits);
    }
}